// KuwaharaBlock_9131100471783
// MI455X (gfx1250) — compile-verified
//
#include <hip/hip_runtime.h>
#include <cstdint>

#define TILE  32
#define R4    40            // TILE + 2*4 halo (luma/pool region)
#define R2    36            // TILE + 2*2 halo (tap/weight region)
#define PSTR  41            // float4 row stride for pixel array (conflict-free)
#define LSTR  41
#define HSTR  37
#define IMG_H 1024
#define IMG_W 1024
#define HW    (IMG_H * IMG_W)
#define NB    4

// ---- gfx1250 async global->LDS path (guarded; falls back to reg copy) ----
#if __has_builtin(__builtin_amdgcn_global_load_async_to_lds_b32)
#define HAVE_ASYNC_LDS 1
#else
#define HAVE_ASYNC_LDS 0
#endif

#if HAVE_ASYNC_LDS
typedef __attribute__((address_space(1))) int as1_int;  // global
typedef __attribute__((address_space(3))) int as3_int;  // LDS
#endif

__device__ __forceinline__ void ld_async_b32(const float* gsrc, float* ldst) {
#if HAVE_ASYNC_LDS
    // Generic->AS3: on AMDGPU the low 32 bits of a generic LDS address are the
    // LDS offset (aperture-based layout), so the truncating cast is exact.
    __builtin_amdgcn_global_load_async_to_lds_b32(
        (as1_int*)(uintptr_t)gsrc,
        (as3_int*)(uint32_t)(uintptr_t)ldst,
        0, 0);
#else
    *ldst = *gsrc;
#endif
}

__device__ __forceinline__ void wait_async_all() {
#if HAVE_ASYNC_LDS
#if __has_builtin(__builtin_amdgcn_s_wait_asynccnt)
    __builtin_amdgcn_s_wait_asynccnt(0);
#else
    asm volatile("s_wait_asynccnt 0" ::: "memory");
#endif
#endif
}

// exp(a) = exp2(a*log2e); constants fold the 64 and /25 factors.
#define C_WGT (-92.33248262f)   // -64*log2(e)
#define C_SPA (-3.6932993047f)  // -64*log2(e)/25

__global__ __launch_bounds__(256)
void kuwahara5_kernel(const float* __restrict__ inp, float* __restrict__ out) {
    __shared__ __align__(16) float s_pix[R4 * PSTR * 4]; // (r,g,b,k->weight)
    __shared__ float s_lum[R4 * LSTR];
    __shared__ float s_lm2[R4 * LSTR];
    __shared__ float s_hs [R4 * HSTR];
    __shared__ float s_hq [R4 * HSTR];

    const int tid = threadIdx.x;
    const int tx0 = blockIdx.x * TILE;
    const int ty0 = blockIdx.y * TILE;
    const float* bin = inp + (size_t)blockIdx.z * 4 * HW;

    // ---- Stage 0: async-stage 40x40 halo-4 region, 4 channels, edge-clamped
    for (int i = tid; i < R4 * R4; i += 256) {
        const int ly = i / R4, lx = i - ly * R4;
        const int gy = ty0 + ly - 4, gx = tx0 + lx - 4;
        const int cy = min(max(gy, 0), IMG_H - 1);
        const int cx = min(max(gx, 0), IMG_W - 1);
        const size_t g = (size_t)cy * IMG_W + cx;
        float* slot = &s_pix[(ly * PSTR + lx) * 4];
        ld_async_b32(bin + g,            slot + 0);
        ld_async_b32(bin + HW + g,       slot + 1);
        ld_async_b32(bin + 2 * HW + g,   slot + 2);
        ld_async_b32(bin + 3 * HW + g,   slot + 3);
    }
    wait_async_all();
    __syncthreads();

    // Save this thread's 4 center k values before .w is overwritten by weights.
    const int sy  = tid >> 3;        // 0..31 (row in tile)
    const int sx0 = (tid & 7) << 2;  // 0,4,...,28 (strip of 4 columns)
    float kc[4];
#pragma unroll
    for (int p = 0; p < 4; ++p)
        kc[p] = s_pix[((sy + 4) * PSTR + (sx0 + p + 4)) * 4 + 3];

    // ---- Stage 1: luma & luma^2 (zero outside image: reduce_window zero-pad)
    for (int i = tid; i < R4 * R4; i += 256) {
        const int ly = i / R4, lx = i - ly * R4;
        const int gy = ty0 + ly - 4, gx = tx0 + lx - 4;
        const bool inim = (gy >= 0) & (gy < IMG_H) & (gx >= 0) & (gx < IMG_W);
        const float4 p4 = *reinterpret_cast<const float4*>(&s_pix[(ly * PSTR + lx) * 4]);
        const float l = inim ? (0.2126f * p4.x + 0.7152f * p4.y + 0.0722f * p4.z) : 0.0f;
        s_lum[ly * LSTR + lx] = l;
        s_lm2[ly * LSTR + lx] = l * l;
    }
    __syncthreads();

    // ---- Stage 2: horizontal 5-sums over 40 rows x 36 cols
    for (int i = tid; i < R4 * R2; i += 256) {
        const int ly = i / R2, xc = i - ly * R2;   // xc = x+2, x in [-2,34)
        const int base = ly * LSTR + xc + 2;       // lum col index (x+4)
        float s = 0.f, qq = 0.f;
#pragma unroll
        for (int dx = -2; dx <= 2; ++dx) { s += s_lum[base + dx]; qq += s_lm2[base + dx]; }
        s_hs[ly * HSTR + xc] = s;
        s_hq[ly * HSTR + xc] = qq;
    }
    __syncthreads();

    // ---- Stage 3: vertical 5-sum (on the fly) -> variance -> weight into .w
    //      Edge-pad of the weight map handled by clamped-coordinate gather.
    for (int i = tid; i < R2 * R2; i += 256) {
        const int wy = i / R2, wx = i - wy * R2;   // region pos y=wy-2, x=wx-2
        const int gy = ty0 + wy - 2, gx = tx0 + wx - 2;
        const int qy = min(max(gy, 0), IMG_H - 1) - ty0;  // in [-2, TILE+1]
        const int qx = min(max(gx, 0), IMG_W - 1) - tx0;
        const int hb = (qy + 4) * HSTR + (qx + 2);
        float sm = 0.f, sq = 0.f;
#pragma unroll
        for (int dy = -2; dy <= 2; ++dy) { sm += s_hs[hb + dy * HSTR]; sq += s_hq[hb + dy * HSTR]; }
        const float mean = sm * 0.04f;
        const float msq  = sq * 0.04f;
        const float var  = fabsf(msq - mean * mean);
        const int   pi   = ((wy + 2) * PSTR + (wx + 2)) * 4;
        const float kq   = s_pix[pi + 3];                    // k at clamped pos
        s_pix[pi + 3] = __builtin_amdgcn_exp2f(var * kq * C_WGT);
    }
    __syncthreads();

    // ---- Stage 4: 25-tap weighted accumulation, 1x4 strip per thread
    float f1[4], f2[4], f4p[4], f5[4], f8[4], cb[4];
#pragma unroll
    for (int p = 0; p < 4; ++p) {
        const float k = kc[p];
        const float f = __builtin_amdgcn_exp2f((1.0f - k) * C_SPA); // exp(-invwidth/25)
        f1[p] = f; f2[p] = f * f; f4p[p] = f2[p] * f2[p];
        f5[p] = f4p[p] * f; f8[p] = f4p[p] * f4p[p];
        cb[p] = 16.0f + k * (0.001f - 16.0f);
    }
    float aR[4] = {0,0,0,0}, aG[4] = {0,0,0,0}, aB[4] = {0,0,0,0}, aW[4] = {0,0,0,0};
    float cR[4], cG[4], cB[4];
#pragma unroll
    for (int dy = -2; dy <= 2; ++dy) {
        float4 q[8];
        const int rb = ((sy + dy + 4) * PSTR + (sx0 + 2)) * 4; // cols sx0-2..sx0+5
#pragma unroll
        for (int j = 0; j < 8; ++j)
            q[j] = *reinterpret_cast<const float4*>(&s_pix[rb + j * 4]);
        if (dy == 0) {
#pragma unroll
            for (int p = 0; p < 4; ++p) { cR[p] = q[p+2].x; cG[p] = q[p+2].y; cB[p] = q[p+2].z; }
        }
#pragma unroll
        for (int p = 0; p < 4; ++p) {
#pragma unroll
            for (int dx = -2; dx <= 2; ++dx) {
                const int s = dy * dy + dx * dx;           // in {0,1,2,4,5,8}
                const float4 v = q[p + dx + 2];
                const float sf = (s == 0) ? 1.0f :
                                 (s == 1) ? f1[p] :
                                 (s == 2) ? f2[p] :
                                 (s == 4) ? f4p[p] :
                                 (s == 5) ? f5[p] : f8[p];
                float w = v.w * sf;
                if (s == 0) w += cb[p];
                aR[p] = fmaf(v.x, w, aR[p]);
                aG[p] = fmaf(v.y, w, aG[p]);
                aB[p] = fmaf(v.z, w, aB[p]);
                aW[p] += w;
            }
        }
    }

    // ---- Final blend + vectorized planar stores
    float* bout = out + (size_t)blockIdx.z * 3 * HW;
    const size_t gbase = (size_t)(ty0 + sy) * IMG_W + (tx0 + sx0);
    float oR[4], oG[4], oB[4];
#pragma unroll
    for (int p = 0; p < 4; ++p) {
        const float inv = 1.0f / aW[p];
        const float k = kc[p];
        oR[p] = cR[p] + k * (aR[p] * inv - cR[p]);
        oG[p] = cG[p] + k * (aG[p] * inv - cG[p]);
        oB[p] = cB[p] + k * (aB[p] * inv - cB[p]);
    }
    *reinterpret_cast<float4*>(&bout[gbase])          = make_float4(oR[0], oR[1], oR[2], oR[3]);
    *reinterpret_cast<float4*>(&bout[HW + gbase])     = make_float4(oG[0], oG[1], oG[2], oG[3]);
    *reinterpret_cast<float4*>(&bout[2 * HW + gbase]) = make_float4(oB[0], oB[1], oB[2], oB[3]);
}

extern "C" void kernel_launch(void* const* d_in, const int* in_sizes, int n_in,
                              void* d_out, int out_size, void* d_ws, size_t ws_size,
                              hipStream_t stream) {
    (void)in_sizes; (void)n_in; (void)d_ws; (void)ws_size; (void)out_size;
    const float* inp = (const float*)d_in[0];   // (4,4,1024,1024) f32
    float* out = (float*)d_out;                 // (4,3,1024,1024) f32
    dim3 grid(IMG_W / TILE, IMG_H / TILE, NB);  // 32 x 32 x 4 tiles
    kuwahara5_kernel<<<grid, dim3(256), 0, stream>>>(inp, out);
}